// CrossAttMultiplexer_24266565222543
// MI455X (gfx1250) — compile-verified
//
#include <hip/hip_runtime.h>
#include <hip/hip_bf16.h>
#include <stdint.h>

// out = WV[0] * s  (the reference attention collapses algebraically; see analysis)
//
// Per wave: async-copy one 16x16 f32 tile (1KB) global->LDS, then compute
// D = (WV * I16) x S via 4x V_WMMA_F32_16X16X4_F32, store D.

typedef __attribute__((ext_vector_type(2))) float v2f;
typedef __attribute__((ext_vector_type(8))) float v8f;

#define WAVES_PER_BLOCK 8
#define TILE_ELEMS      256   // 16x16 f32 per wave
#define BLOCK_ELEMS     (WAVES_PER_BLOCK * TILE_ELEMS)  // 2048

__global__ __launch_bounds__(256) void scale_tile_wmma(const float* __restrict__ s,
                                                       const float* __restrict__ WV,
                                                       float* __restrict__ out) {
  __shared__ __align__(16) float lds[WAVES_PER_BLOCK * TILE_ELEMS];

  const int lane = threadIdx.x & 31;
  const int wid  = threadIdx.x >> 5;
  const int half = lane >> 4;   // 0: lanes 0-15, 1: lanes 16-31
  const int col  = lane & 15;

  const long long tileBase = ((long long)blockIdx.x * WAVES_PER_BLOCK + wid) * TILE_ELEMS;
  const float* src = s + tileBase;
  float* dst = out + tileBase;

  // --- async global -> LDS staging (ASYNCcnt path), 2 x 512B per wave ---
  float* lw = &lds[wid * TILE_ELEMS];
  unsigned lds0 = (unsigned)(uintptr_t)(lw + lane * 4);   // 16B per lane
  const float* g0 = src + lane * 4;
  asm volatile("global_load_async_to_lds_b128 %0, %1, off"
               :: "v"(lds0), "v"(g0) : "memory");
  asm volatile("global_load_async_to_lds_b128 %0, %1, off"
               :: "v"(lds0 + 512u), "v"(g0 + 128) : "memory");

  const float c = WV[0];  // overlap scalar load with async copy

  asm volatile("s_wait_asynccnt 0" ::: "memory");

  // --- D = (c * I16) x S via 4 accumulating 16x16x4 f32 WMMAs ---
  // A 16x4 layout: VGPR0 holds K=2*half, VGPR1 holds K=2*half+1, M = lane&15.
  // B 4x16 layout mirrors A: VGPR0 row K=2*half, VGPR1 row K=2*half+1, N = lane&15.
  v8f acc = {};
#pragma unroll
  for (int kb = 0; kb < 4; ++kb) {
    const int r0 = 4 * kb + 2 * half;   // K index of this lane's VGPR0 slot
    v2f b;
    b.x = lw[(r0    ) * 16 + col];      // S row r0,   col n   (bank-conflict-free)
    b.y = lw[(r0 + 1) * 16 + col];      // S row r0+1, col n
    v2f a;
    a.x = (col == r0    ) ? c : 0.0f;   // (c*I)[M][K]
    a.y = (col == r0 + 1) ? c : 0.0f;
    acc = __builtin_amdgcn_wmma_f32_16x16x4_f32(
        /*neg_a=*/false, a, /*neg_b=*/false, b,
        /*c_mod=*/(short)0, acc, /*reuse_a=*/false, /*reuse_b=*/false);
  }

  // D layout: VGPR r -> row (r + 8*half), col = lane&15
#pragma unroll
  for (int r = 0; r < 8; ++r) {
    dst[(r + 8 * half) * 16 + col] = acc[r];
  }
}

// Tail: plain elementwise scale for any remainder (not hit for the given shapes).
__global__ void scale_tail(const float* __restrict__ s, const float* __restrict__ WV,
                           float* __restrict__ out, long long start, long long total) {
  long long i = start + blockIdx.x * (long long)blockDim.x + threadIdx.x;
  if (i < total) out[i] = WV[0] * s[i];
}

extern "C" void kernel_launch(void* const* d_in, const int* in_sizes, int n_in,
                              void* d_out, int out_size, void* d_ws, size_t ws_size,
                              hipStream_t stream) {
  // inputs: 0=x (unused), 1=s, 2=WQ (unused), 3=WK (unused), 4=WV
  const float* s  = (const float*)d_in[1];
  const float* WV = (const float*)d_in[4];
  float* out = (float*)d_out;

  const long long total = (long long)in_sizes[1];       // 4*64*64*96 = 1,572,864
  const long long nblocks = total / BLOCK_ELEMS;        // 768 exact for these shapes

  if (nblocks > 0) {
    scale_tile_wmma<<<dim3((unsigned)nblocks), dim3(256), 0, stream>>>(s, WV, out);
  }
  const long long done = nblocks * BLOCK_ELEMS;
  const long long rem = total - done;
  if (rem > 0) {
    unsigned tb = (unsigned)((rem + 255) / 256);
    scale_tail<<<dim3(tb), dim3(256), 0, stream>>>(s, WV, out, done, total);
  }
}